// DeeperGCN_38190849196651
// MI455X (gfx1250) — compile-verified
//
#include <hip/hip_runtime.h>
#include <hip/hip_bf16.h>
#include <math.h>

// ---------------------------------------------------------------------------
// Sizes (fixed by the reference problem)
// ---------------------------------------------------------------------------
#define GN  50000
#define GE  1250000
#define GFX 128
#define GFE 8
#define GH  64
#define GC  112
#define GL  28

typedef __attribute__((ext_vector_type(2))) float v2f;
typedef __attribute__((ext_vector_type(8))) float v8f;

// ---------------------------------------------------------------------------
// WMMA f32 GEMM:  C[M x Nc] = A[M x K] @ B[K x Nc] + bias  (optionally += C)
// One block = one 16-row stripe of A; 4 waves cover the Nc/16 column tiles.
// Uses V_WMMA_F32_16X16X4_F32 (f32 matrix core path on gfx1250).
// A-frag layout (ISA 7.12.2): lanes 0-15 hold K=k,k+1; lanes 16-31 K=k+2,k+3.
// C/D layout: VGPR r -> M = r + 8*(lane>=16), N = lane&15.
// ---------------------------------------------------------------------------
__global__ void __launch_bounds__(128)
gemm16_wmma(const float* __restrict__ A, const float* __restrict__ B,
            const float* __restrict__ bias, float* __restrict__ C,
            int M, int Nc, int K, int accum) {
    __shared__ float sA[16 * 129];              // 16 rows, padded stride (K<=128)
    const int m0   = blockIdx.x << 4;
    const int tid  = threadIdx.x;
    const int strd = K + 1;

    // Stage the 16 x K stripe of A (coalesced global reads).
    for (int i = tid; i < 16 * K; i += 128) {
        int r = i / K, c = i - r * K;
        sA[r * strd + c] = A[(size_t)(m0 + r) * (size_t)K + c];
    }
    __syncthreads();

    const int wave  = tid >> 5;
    const int lane  = tid & 31;
    const int lhalf = lane >> 4;                // 0 or 1
    const int l16   = lane & 15;
    const int ntile = Nc >> 4;

    for (int nt = wave; nt < ntile; nt += 4) { // wave-uniform loop (EXEC stays full)
        const int n0 = nt << 4;
        float bv = bias ? bias[n0 + l16] : 0.0f;
        v8f acc;
        #pragma unroll
        for (int r = 0; r < 8; ++r) acc[r] = bv;

        for (int k = 0; k < K; k += 4) {
            const int kk = k + 2 * lhalf;
            v2f a, b;
            a[0] = sA[l16 * strd + kk];
            a[1] = sA[l16 * strd + kk + 1];
            b[0] = B[(size_t)kk       * (size_t)Nc + n0 + l16];   // L2-resident
            b[1] = B[(size_t)(kk + 1) * (size_t)Nc + n0 + l16];
            acc = __builtin_amdgcn_wmma_f32_16x16x4_f32(
                    false, a, false, b, (short)0, acc, false, false);
        }

        #pragma unroll
        for (int r = 0; r < 8; ++r) {
            size_t row = (size_t)(m0 + r + 8 * lhalf);
            size_t idx = row * (size_t)Nc + (size_t)(n0 + l16);
            float v = acc[r];
            C[idx] = accum ? (C[idx] + v) : v;
        }
    }
}

// ---------------------------------------------------------------------------
// Float atomic max via ordered-int trick (global_atomic_max_i32 / min_u32).
// Works for mixed signs when the slot is initialized to -inf bits.
// ---------------------------------------------------------------------------
__device__ __forceinline__ void atomicMaxFloat(float* addr, float val) {
    if (val >= 0.0f)
        atomicMax((int*)addr, __float_as_int(val));
    else
        atomicMin((unsigned int*)addr, __float_as_uint(val));
}

// m <- -inf, den <- 0, num <- 0
__global__ void init_seg(float* __restrict__ m, float* __restrict__ den,
                         float* __restrict__ num, size_t n) {
    size_t i = (size_t)blockIdx.x * blockDim.x + threadIdx.x;
    if (i < n) {
        m[i]   = -__builtin_inff();
        den[i] = 0.0f;
        num[i] = 0.0f;
    }
}

// Pass 1: segment-max of logits over dst.  16 threads/edge, float4 per thread.
__global__ void edge_pass1(const float* __restrict__ xin, const float* __restrict__ ea,
                           const int* __restrict__ src, const int* __restrict__ dst,
                           const float* __restrict__ tptr, int layer,
                           float* __restrict__ m, int E) {
    int idx = blockIdx.x * blockDim.x + threadIdx.x;
    if (idx >= E * 16) return;
    int e = idx >> 4;
    int q = (idx & 15) << 2;
    float t = tptr[layer];
    int s = src[e], d = dst[e];
    float4 xa = *(const float4*)(xin + (size_t)s * GH + q);
    float4 eb = *(const float4*)(ea  + (size_t)e * GH + q);
    float* mp = m + (size_t)d * GH + q;
    const float* xp = &xa.x;
    const float* ep = &eb.x;
    #pragma unroll
    for (int j = 0; j < 4; ++j) {
        float msg   = fmaxf(xp[j] + ep[j], 0.0f) + 1e-7f;
        float logit = msg * t;
        atomicMaxFloat(mp + j, logit);
    }
}

// Replace non-finite maxima (empty segments) with 0.
__global__ void finalize_max(float* __restrict__ m, size_t n) {
    size_t i = (size_t)blockIdx.x * blockDim.x + threadIdx.x;
    if (i < n) {
        float v = m[i];
        if (!__builtin_isfinite(v)) m[i] = 0.0f;
    }
}

// Pass 2: den += exp(logit - m[dst]); num += msg * exp(...)
__global__ void edge_pass2(const float* __restrict__ xin, const float* __restrict__ ea,
                           const int* __restrict__ src, const int* __restrict__ dst,
                           const float* __restrict__ tptr, int layer,
                           const float* __restrict__ m,
                           float* __restrict__ den, float* __restrict__ num, int E) {
    int idx = blockIdx.x * blockDim.x + threadIdx.x;
    if (idx >= E * 16) return;
    int e = idx >> 4;
    int q = (idx & 15) << 2;
    float t = tptr[layer];
    int s = src[e], d = dst[e];
    float4 xa = *(const float4*)(xin + (size_t)s * GH + q);
    float4 eb = *(const float4*)(ea  + (size_t)e * GH + q);
    float4 mm = *(const float4*)(m   + (size_t)d * GH + q);
    float* dn = den + (size_t)d * GH + q;
    float* nm = num + (size_t)d * GH + q;
    const float* xp = &xa.x;
    const float* ep = &eb.x;
    const float* mp = &mm.x;
    #pragma unroll
    for (int j = 0; j < 4; ++j) {
        float msg   = fmaxf(xp[j] + ep[j], 0.0f) + 1e-7f;
        float logit = msg * t;
        float ex    = expf(logit - mp[j]);
        atomicAdd(dn + j, ex);
        atomicAdd(nm + j, msg * ex);
    }
}

// out = num / max(den,>0 else 1) + xin
__global__ void node_agg(const float* __restrict__ num, const float* __restrict__ den,
                         const float* __restrict__ xin, float* __restrict__ out, size_t n) {
    size_t i = (size_t)blockIdx.x * blockDim.x + threadIdx.x;
    if (i < n) {
        float dv = den[i];
        out[i] = num[i] / (dv > 0.0f ? dv : 1.0f) + xin[i];
    }
}

// y = relu(LayerNorm(x) * g + b); one wave32 per row; D in {64,128}
__global__ void ln_relu(const float* __restrict__ x, const float* __restrict__ g,
                        const float* __restrict__ b, float* __restrict__ y,
                        int D, int rows) {
    int wid  = (int)((blockIdx.x * (size_t)blockDim.x + threadIdx.x) >> 5);
    int lane = threadIdx.x & 31;
    if (wid >= rows) return;
    const float* xr = x + (size_t)wid * D;
    float v[4];
    int per = D >> 5;                      // elems per lane: 2 or 4
    float s = 0.0f;
    for (int j = 0; j < per; ++j) { v[j] = xr[lane + 32 * j]; s += v[j]; }
    for (int off = 16; off; off >>= 1) s += __shfl_xor(s, off, 32);
    float mu = s / (float)D;
    float vs = 0.0f;
    for (int j = 0; j < per; ++j) { float d = v[j] - mu; vs += d * d; }
    for (int off = 16; off; off >>= 1) vs += __shfl_xor(vs, off, 32);
    float inv = rsqrtf(vs / (float)D + 1e-5f);
    float* yr = y + (size_t)wid * D;
    for (int j = 0; j < per; ++j) {
        int c = lane + 32 * j;
        float o = (v[j] - mu) * inv * g[c] + b[c];
        yr[c] = fmaxf(o, 0.0f);
    }
}

// ---------------------------------------------------------------------------
// Host orchestration (graph-capture safe: only kernel launches on `stream`)
// ---------------------------------------------------------------------------
extern "C" void kernel_launch(void* const* d_in, const int* in_sizes, int n_in,
                              void* d_out, int out_size, void* d_ws, size_t ws_size,
                              hipStream_t stream) {
    const float* x          = (const float*)d_in[0];
    const float* edge_attr  = (const float*)d_in[1];
    const int*   edge_index = (const int*)  d_in[2];
    const float* W_node     = (const float*)d_in[3];
    const float* b_node     = (const float*)d_in[4];
    const float* W_edge     = (const float*)d_in[5];
    const float* b_edge     = (const float*)d_in[6];
    const float* conv_W1    = (const float*)d_in[7];   // L x H x 2H
    const float* conv_b1    = (const float*)d_in[8];   // L x 2H
    const float* conv_ln_g  = (const float*)d_in[9];
    const float* conv_ln_b  = (const float*)d_in[10];
    const float* conv_W2    = (const float*)d_in[11];  // L x 2H x H
    const float* conv_b2    = (const float*)d_in[12];
    const float* conv_t     = (const float*)d_in[13];  // L
    const float* layer_ln_g = (const float*)d_in[14];  // L x H
    const float* layer_ln_b = (const float*)d_in[15];
    const float* W_out      = (const float*)d_in[16];  // H x C
    const float* b_out      = (const float*)d_in[17];
    float* out = (float*)d_out;

    const int Nn = GN, Ee = GE, H = GH;
    const int* src = edge_index;
    const int* dst = edge_index + Ee;

    // workspace layout (floats)
    float* ws = (float*)d_ws;
    size_t o = 0;
    float* ea    = ws + o; o += (size_t)Ee * H;   // 320 MB
    float* carry = ws + o; o += (size_t)Nn * H;
    float* zbuf  = ws + o; o += (size_t)Nn * H;
    float* mbuf  = ws + o; o += (size_t)Nn * H;
    float* den   = ws + o; o += (size_t)Nn * H;
    float* num   = ws + o; o += (size_t)Nn * H;
    float* outb  = ws + o; o += (size_t)Nn * H;
    float* t1    = mbuf;   // N x 2H, aliases mbuf+den (dead after node_agg)

    const size_t NH = (size_t)Nn * H;
    const int nh_blocks = (int)((NH + 255) / 256);
    const int eh_blocks = (Ee * 16 + 255) / 256;
    const int ln_blocks = (Nn + 7) / 8;           // 8 wave32 rows per 256-thr block

    auto gemm = [&](const float* A, const float* B, const float* bias, float* Cp,
                    int M, int Nc, int K, int accum) {
        gemm16_wmma<<<dim3(M / 16), dim3(128), 0, stream>>>(A, B, bias, Cp, M, Nc, K, accum);
    };

    // h = x @ W_node + b_node ; ea = edge_attr @ W_edge + b_edge
    gemm(x,         W_node, b_node, carry, Nn, H, GFX, 0);
    gemm(edge_attr, W_edge, b_edge, ea,    Ee, H, GFE, 0);

    for (int l = 0; l < GL; ++l) {
        const float* xin;
        if (l == 0) {
            xin = carry;
        } else {
            ln_relu<<<ln_blocks, 256, 0, stream>>>(carry, layer_ln_g + l * H,
                                                   layer_ln_b + l * H, zbuf, H, Nn);
            xin = zbuf;
        }
        init_seg<<<nh_blocks, 256, 0, stream>>>(mbuf, den, num, NH);
        edge_pass1<<<eh_blocks, 256, 0, stream>>>(xin, ea, src, dst, conv_t, l, mbuf, Ee);
        finalize_max<<<nh_blocks, 256, 0, stream>>>(mbuf, NH);
        edge_pass2<<<eh_blocks, 256, 0, stream>>>(xin, ea, src, dst, conv_t, l,
                                                  mbuf, den, num, Ee);
        node_agg<<<nh_blocks, 256, 0, stream>>>(num, den, xin, outb, NH);

        // t1 = relu(LN(outb @ W1 + b1)) ; genconv out accumulated into carry
        gemm(outb, conv_W1 + (size_t)l * H * 2 * H, conv_b1 + l * 2 * H, t1, Nn, 2 * H, H, 0);
        ln_relu<<<ln_blocks, 256, 0, stream>>>(t1, conv_ln_g + l * 2 * H,
                                               conv_ln_b + l * 2 * H, t1, 2 * H, Nn);
        gemm(t1, conv_W2 + (size_t)l * 2 * H * H, conv_b2 + l * H, carry, Nn, H, 2 * H,
             l == 0 ? 0 : 1);
    }

    // final: out = relu(LN(carry, g0, b0)) @ W_out + b_out
    ln_relu<<<ln_blocks, 256, 0, stream>>>(carry, layer_ln_g, layer_ln_b, zbuf, H, Nn);
    gemm(zbuf, W_out, b_out, out, Nn, GC, H, 0);
}